// HANLayer_23768349016455
// MI455X (gfx1250) — compile-verified
//
#include <hip/hip_runtime.h>
#include <cmath>

#define H  8
#define D  32
#define HD 256
#define NEG_SLOPE 0.2f

typedef float v2f __attribute__((ext_vector_type(2)));
typedef float v8f __attribute__((ext_vector_type(8)));

// ---------------------------------------------------------------------------
// GEMM: C[M,N] = A[M,K] * B[K,N], fp32, via V_WMMA_F32_16X16X4_F32.
// N%16==0, K%4==0, M%16==0 (all true here).
// One wave computes a 64x16 strip of C (4 stacked 16x16 tiles) so the B
// fragment is loaded once per k-step and reused by 4 WMMAs:
//   per 4 WMMAs: 4x global_load_b64 (A) + 2x global_load_b32 (B).
// Block = 128 threads = 4 waves covering 64 consecutive columns.
// Grid = (N/64, ceil(M/64)).
//
// VGPR layouts (CDNA5 ISA 7.12.2):
//   A 16x4 f32: lanes 0-15 hold {K=k+0,k+1}, lanes 16-31 hold {K=k+2,k+3},
//               M = lane&15.
//   B 4x16 f32: mirrored - lanes 0-15 rows k+0/k+1, lanes 16-31 rows k+2/k+3,
//               N = lane&15.
//   C 16x16 f32: VGPR i -> M = i + 8*(lane>>4), N = lane&15.
// ---------------------------------------------------------------------------
__global__ void gemm_f32_wmma(const float* __restrict__ A,
                              const float* __restrict__ B,
                              float* __restrict__ C,
                              int M, int N, int K) {
  const int lane = threadIdx.x & 31;
  const int wave = threadIdx.x >> 5;
  const int hl   = lane >> 4;      // which half of the wave
  const int l16  = lane & 15;
  const int n0   = (blockIdx.x * 4 + wave) * 16;
  const int mb   = blockIdx.y * 64;
  if (n0 >= N || mb >= M) return;

  // Per-tile validity: tiles are 16-aligned and M%16==0, so each 16-row tile
  // is entirely valid or entirely OOB.  Clamp OOB tiles' loads to row l16
  // (wave-uniform predicate; EXEC stays all-ones for WMMA).
  bool valid[4];
  const float* arow[4];
#pragma unroll
  for (int t = 0; t < 4; ++t) {
    int mt   = mb + 16 * t;
    valid[t] = (mt < M);
    int r    = (valid[t] ? mt : 0) + l16;
    arow[t]  = A + (size_t)r * K + 2 * hl;
  }

  const float* bcol = B + (size_t)(2 * hl) * N + n0 + l16;

  v8f c0 = {}, c1 = {}, c2 = {}, c3 = {};
  for (int k = 0; k < K; k += 4) {
    v2f b;                                     // shared across 4 m-tiles
    b.x = bcol[(size_t)k * N];                 // B[k+2h  ][n]
    b.y = bcol[(size_t)(k + 1) * N];           // B[k+2h+1][n]
    v2f a0 = *(const v2f*)(arow[0] + k);       // A[m][k+2h..k+2h+1]
    v2f a1 = *(const v2f*)(arow[1] + k);
    v2f a2 = *(const v2f*)(arow[2] + k);
    v2f a3 = *(const v2f*)(arow[3] + k);
    c0 = __builtin_amdgcn_wmma_f32_16x16x4_f32(false, a0, false, b, (short)0, c0, false, false);
    c1 = __builtin_amdgcn_wmma_f32_16x16x4_f32(false, a1, false, b, (short)0, c1, false, false);
    c2 = __builtin_amdgcn_wmma_f32_16x16x4_f32(false, a2, false, b, (short)0, c2, false, false);
    c3 = __builtin_amdgcn_wmma_f32_16x16x4_f32(false, a3, false, b, (short)0, c3, false, false);
  }

  v8f acc[4] = {c0, c1, c2, c3};
#pragma unroll
  for (int t = 0; t < 4; ++t) {
    if (!valid[t]) continue;                   // wave-uniform
    float* crow = C + (size_t)(mb + 16 * t + 8 * hl) * N + n0 + l16;
#pragma unroll
    for (int i = 0; i < 8; ++i)
      crow[(size_t)i * N] = acc[t][i];
  }
}

// ---------------------------------------------------------------------------
// e[n,h] = sum_d f[n, h*32+d] * att[h*32+d]
// ---------------------------------------------------------------------------
__global__ void att_score(const float* __restrict__ f,
                          const float* __restrict__ att,
                          float* __restrict__ e, int n_nodes) {
  int i = blockIdx.x * blockDim.x + threadIdx.x;
  if (i >= n_nodes * H) return;
  int node = i >> 3, h = i & 7;
  const float* fp = f + (size_t)node * HD + h * D;
  const float* ap = att + h * D;
  float s = 0.f;
#pragma unroll
  for (int d = 0; d < D; d += 4) {
    float4 fv = *(const float4*)(fp + d);
    float4 av = *(const float4*)(ap + d);
    s += fv.x * av.x + fv.y * av.y + fv.z * av.z + fv.w * av.w;
  }
  e[i] = s;
}

__global__ void fill_f32(float* __restrict__ p, float v, int n) {
  int i = blockIdx.x * blockDim.x + threadIdx.x;
  if (i < n) p[i] = v;
}

__device__ __forceinline__ float leaky(float v) {
  return v > 0.f ? v : v * NEG_SLOPE;
}

// float atomic max via monotone int mapping (valid over the whole float order)
__device__ __forceinline__ void atomicMaxF(float* addr, float val) {
  int iv = __float_as_int(val);
  if (iv >= 0) atomicMax((int*)addr, iv);
  else         atomicMin((unsigned int*)addr, (unsigned int)iv);
}

// ---------------------------------------------------------------------------
// Pass 1: per-dst running max of leaky_relu(el[src]+er[dst]);  thread=(edge,h)
// ---------------------------------------------------------------------------
__global__ void edge_max(const int* __restrict__ src, const int* __restrict__ dst,
                         const float* __restrict__ el, const float* __restrict__ er,
                         float* __restrict__ emax, int E) {
  int i = blockIdx.x * blockDim.x + threadIdx.x;
  if (i >= E * H) return;
  int e = i >> 3, h = i & 7;
  int s = src[e], d0 = dst[e];
  float v = leaky(el[s * H + h] + er[d0 * H + h]);
  atomicMaxF(&emax[d0 * H + h], v);
}

// ---------------------------------------------------------------------------
// Pass 2: denom[dst,h] += exp(e - emax[dst,h])
// ---------------------------------------------------------------------------
__global__ void edge_expsum(const int* __restrict__ src, const int* __restrict__ dst,
                            const float* __restrict__ el, const float* __restrict__ er,
                            const float* __restrict__ emax, float* __restrict__ denom,
                            int E) {
  int i = blockIdx.x * blockDim.x + threadIdx.x;
  if (i >= E * H) return;
  int e = i >> 3, h = i & 7;
  int s = src[e], d0 = dst[e];
  float v = leaky(el[s * H + h] + er[d0 * H + h]);
  atomicAdd(&denom[d0 * H + h], expf(v - emax[d0 * H + h]));
}

// ---------------------------------------------------------------------------
// Pass 3: out[dst,:] += fs[src,:] * a   (a = softmax coefficient per head)
// 32 lanes per edge; lane handles 8 contiguous channels (one head spans 4 lanes)
// ---------------------------------------------------------------------------
__global__ void edge_scatter(const int* __restrict__ src, const int* __restrict__ dst,
                             const float* __restrict__ fs,
                             const float* __restrict__ el, const float* __restrict__ er,
                             const float* __restrict__ emax, const float* __restrict__ denom,
                             float* __restrict__ out, int E) {
  int i = blockIdx.x * blockDim.x + threadIdx.x;
  int e = i >> 5;
  if (e >= E) return;
  int lane = i & 31;
  int h = lane >> 2;                       // 32 channels per head / 8 per lane
  int s = src[e], d0 = dst[e];
  float v = leaky(el[s * H + h] + er[d0 * H + h]);
  float w = expf(v - emax[d0 * H + h]);
  float a = w / fmaxf(denom[d0 * H + h], 1e-9f);

  const float* fp = fs + (size_t)s * HD + lane * 8;
  float*       op = out + (size_t)d0 * HD + lane * 8;
  float4 f0 = *(const float4*)(fp);
  float4 f1 = *(const float4*)(fp + 4);
  atomicAdd(op + 0, f0.x * a);
  atomicAdd(op + 1, f0.y * a);
  atomicAdd(op + 2, f0.z * a);
  atomicAdd(op + 3, f0.w * a);
  atomicAdd(op + 4, f1.x * a);
  atomicAdd(op + 5, f1.y * a);
  atomicAdd(op + 6, f1.z * a);
  atomicAdd(op + 7, f1.w * a);
}

// ---------------------------------------------------------------------------
// out = elu(out + bias)  (in place)
// ---------------------------------------------------------------------------
__global__ void finalize_elu(float* __restrict__ out, const float* __restrict__ b,
                             int n_nodes) {
  int i = blockIdx.x * blockDim.x + threadIdx.x;
  if (i >= n_nodes * HD) return;
  float v = out[i] + b[i & (HD - 1)];
  out[i] = v > 0.f ? v : expm1f(v);
}

// ---------------------------------------------------------------------------
// Host-side driver for one GAT relation.
// ---------------------------------------------------------------------------
static void run_relation(const float* xs, int Ns, int Ks,
                         const float* xd, int Nd, int Kd,
                         const float* Wsrc, const float* Wdst,
                         const float* al, const float* ar, const float* bias,
                         const int* src, const int* dst, int E,
                         float* bufS, float* bufD,
                         float* el, float* er, float* emax, float* denom,
                         float* out, hipStream_t stream) {
  dim3 blk(128);
  gemm_f32_wmma<<<dim3(HD / 64, (Ns + 63) / 64), blk, 0, stream>>>(xs, Wsrc, bufS, Ns, HD, Ks);
  const float* fd = bufS;                       // self-loop: shared projection
  if (!(xd == xs && Wdst == Wsrc)) {
    gemm_f32_wmma<<<dim3(HD / 64, (Nd + 63) / 64), blk, 0, stream>>>(xd, Wdst, bufD, Nd, HD, Kd);
    fd = bufD;
  }
  att_score<<<(Ns * H + 255) / 256, 256, 0, stream>>>(bufS, al, el, Ns);
  att_score<<<(Nd * H + 255) / 256, 256, 0, stream>>>(fd, ar, er, Nd);
  fill_f32<<<(Nd * H + 255) / 256, 256, 0, stream>>>(emax, -INFINITY, Nd * H);
  fill_f32<<<(Nd * H + 255) / 256, 256, 0, stream>>>(denom, 0.f, Nd * H);
  edge_max   <<<(E * H + 255) / 256, 256, 0, stream>>>(src, dst, el, er, emax, E);
  edge_expsum<<<(E * H + 255) / 256, 256, 0, stream>>>(src, dst, el, er, emax, denom, E);
  edge_scatter<<<((size_t)E * 32 + 255) / 256, 256, 0, stream>>>(
      src, dst, bufS, el, er, emax, denom, out, E);
  finalize_elu<<<(Nd * HD + 255) / 256, 256, 0, stream>>>(out, bias, Nd);
}

extern "C" void kernel_launch(void* const* d_in, const int* in_sizes, int n_in,
                              void* d_out, int out_size, void* d_ws, size_t ws_size,
                              hipStream_t stream) {
  const int DF = 256, DA = 128;
  const float* h_file   = (const float*)d_in[0];
  const float* h_author = (const float*)d_in[1];
  const int NF = in_sizes[0] / DF;   // 40000
  const int NA = in_sizes[1] / DA;   // 10000

  const int* fa_src = (const int*)d_in[2];  const int* fa_dst = (const int*)d_in[3];
  const int* af_src = (const int*)d_in[4];  const int* af_dst = (const int*)d_in[5];
  const int* fs_src = (const int*)d_in[6];  const int* fs_dst = (const int*)d_in[7];
  const int* as_src = (const int*)d_in[8];  const int* as_dst = (const int*)d_in[9];
  const int E_FA = in_sizes[2], E_AF = in_sizes[4], E_FS = in_sizes[6], E_AS = in_sizes[8];

  const float* fa_Wsrc = (const float*)d_in[10];
  const float* fa_Wdst = (const float*)d_in[11];
  const float* fa_al   = (const float*)d_in[12];
  const float* fa_ar   = (const float*)d_in[13];
  const float* fa_b    = (const float*)d_in[14];
  const float* af_Wsrc = (const float*)d_in[15];
  const float* af_Wdst = (const float*)d_in[16];
  const float* af_al   = (const float*)d_in[17];
  const float* af_ar   = (const float*)d_in[18];
  const float* af_b    = (const float*)d_in[19];
  const float* fs_W    = (const float*)d_in[20];
  const float* fs_al   = (const float*)d_in[21];
  const float* fs_ar   = (const float*)d_in[22];
  const float* fs_b    = (const float*)d_in[23];
  const float* as_W    = (const float*)d_in[24];
  const float* as_al   = (const float*)d_in[25];
  const float* as_ar   = (const float*)d_in[26];
  const float* as_b    = (const float*)d_in[27];

  // Workspace layout (sized for the largest relation, reused sequentially)
  float* bufS  = (float*)d_ws;                  // [NF, 256]
  float* bufD  = bufS  + (size_t)NF * HD;       // [NF, 256]
  float* el    = bufD  + (size_t)NF * HD;       // [NF, 8]
  float* er    = el    + (size_t)NF * H;        // [NF, 8]
  float* emax  = er    + (size_t)NF * H;        // [NF, 8]
  float* denom = emax  + (size_t)NF * H;        // [NF, 8]

  // Output slices: (author_fa, file_af, file_self, author_self)
  float* out_author_fa   = (float*)d_out;
  float* out_file_af     = out_author_fa   + (size_t)NA * HD;
  float* out_file_self   = out_file_af     + (size_t)NF * HD;
  float* out_author_self = out_file_self   + (size_t)NF * HD;

  hipMemsetAsync(d_out, 0, (size_t)out_size * sizeof(float), stream);

  // file -> author
  run_relation(h_file, NF, DF, h_author, NA, DA, fa_Wsrc, fa_Wdst,
               fa_al, fa_ar, fa_b, fa_src, fa_dst, E_FA,
               bufS, bufD, el, er, emax, denom, out_author_fa, stream);
  // author -> file
  run_relation(h_author, NA, DA, h_file, NF, DF, af_Wsrc, af_Wdst,
               af_al, af_ar, af_b, af_src, af_dst, E_AF,
               bufS, bufD, el, er, emax, denom, out_file_af, stream);
  // file self-loop (shared projection)
  run_relation(h_file, NF, DF, h_file, NF, DF, fs_W, fs_W,
               fs_al, fs_ar, fs_b, fs_src, fs_dst, E_FS,
               bufS, bufD, el, er, emax, denom, out_file_self, stream);
  // author self-loop (shared projection)
  run_relation(h_author, NA, DA, h_author, NA, DA, as_W, as_W,
               as_al, as_ar, as_b, as_src, as_dst, E_AS,
               bufS, bufD, el, er, emax, denom, out_author_self, stream);
}